// LATTE_25546465476711
// MI455X (gfx1250) — compile-verified
//
#include <hip/hip_runtime.h>

#define H 4
#define C 16
#define D_IN 128
#define D_OUT 64
#define N_AUTHOR 150000
#define N_PAPER 200000
#define N_OUT 50000
#define NEDGE 1000000
#define NEG_SLOPE 0.2f

typedef __attribute__((ext_vector_type(2))) float v2f;
typedef __attribute__((ext_vector_type(8))) float v8f;

__device__ __forceinline__ float lrelu(float x) { return x > 0.f ? x : NEG_SLOPE * x; }

// Order-preserving float<->uint mapping so segment-max can use integer atomicMax.
__device__ __forceinline__ unsigned f2o(float f) {
    unsigned u = __float_as_uint(f);
    return (u & 0x80000000u) ? ~u : (u | 0x80000000u);
}
__device__ __forceinline__ float o2f(unsigned u) {
    return (u & 0x80000000u) ? __uint_as_float(u & 0x7fffffffu) : __uint_as_float(~u);
}
#define ORD_NEG_INF 0x007fffffu  // f2o(-inf)

// ---------------------------------------------------------------------------
// Y[M x 64] = X[M x 128] @ W[128 x 64] + bias, exact fp32 via V_WMMA_F32_16X16X4_F32.
// One wave owns one 16-wide N tile (n0 = (w&3)*16), caches the whole 128x16 B
// panel in 64 VGPRs, then streams 16-row M tiles (32 chained WMMAs, K=128).
// A layout (16x4 f32): lanes 0-15 = M rows, v0/v1 = K0/K1; lanes 16-31 = K2/K3.
// C/D layout: VGPR r, lanes 0-15 -> (M=r, N=lane), lanes 16-31 -> (M=8+r).
// ---------------------------------------------------------------------------
__global__ void gemm_bias_wmma(const float* __restrict__ X, const float* __restrict__ W,
                               const float* __restrict__ bias, float* __restrict__ Y,
                               int mtiles) {
    const int wavesPerBlock = blockDim.x >> 5;
    const int w    = blockIdx.x * wavesPerBlock + (threadIdx.x >> 5);
    const int lane = threadIdx.x & 31;
    const int lm   = lane & 15;           // M (for A) / N (for B,C) index
    const int ko   = (lane >> 4) << 1;    // 0 or 2: K sub-offset within the 4-wide step
    const int n0   = (w & 3) * 16;
    const int totalWaves = gridDim.x * wavesPerBlock;
    const int mstride    = totalWaves >> 2;

    // Preload B panel: b[k] = (W[4k+ko][n0+lm], W[4k+ko+1][n0+lm])
    v2f b[32];
#pragma unroll
    for (int k = 0; k < 32; ++k) {
        const float* wp = W + (size_t)(4 * k + ko) * D_OUT + n0 + lm;
        v2f t; t.x = wp[0]; t.y = wp[D_OUT];
        b[k] = t;
    }
    const float bv = bias[n0 + lm];

    for (int mt = (w >> 2); mt < mtiles; mt += mstride) {
        const float* arow = X + (size_t)(mt * 16 + lm) * D_IN + ko;
        v8f c = {};
#pragma unroll
        for (int k = 0; k < 32; ++k) {
            v2f a = *(const v2f*)(arow + 4 * k);  // (X[row][4k+ko], X[row][4k+ko+1])
            c = __builtin_amdgcn_wmma_f32_16x16x4_f32(false, a, false, b[k],
                                                      (short)0, c, false, false);
        }
        float* yp = Y + (size_t)(mt * 16 + (lane >> 4) * 8) * D_OUT + n0 + lm;
#pragma unroll
        for (int r = 0; r < 8; ++r) yp[(size_t)r * D_OUT] = c[r] + bv;
    }
}

// ---------------------------------------------------------------------------
// Re-init accumulators each call (d_ws is poisoned by the harness).
// ---------------------------------------------------------------------------
__global__ void init_kernel(unsigned* __restrict__ mw, unsigned* __restrict__ mc,
                            float* __restrict__ dw, float* __restrict__ dc,
                            float* __restrict__ aw, float* __restrict__ ac) {
    int i = blockIdx.x * blockDim.x + threadIdx.x;
    if (i < N_OUT * H) { mw[i] = ORD_NEG_INF; mc[i] = ORD_NEG_INF; dw[i] = 0.f; dc[i] = 0.f; }
    if (i < N_OUT * D_OUT) { aw[i] = 0.f; ac[i] = 0.f; }
}

// ---------------------------------------------------------------------------
// Pass 1: per-edge attention logits + segment max.
// 16 lanes cooperate on one edge: lane t reads channels [4t,4t+4) of both rows
// (one coalesced 256B row read per matrix), head h = t/4, 4-lane shfl reduce.
// ---------------------------------------------------------------------------
__global__ void edge_alpha_kernel(const float* __restrict__ lsrc, const float* __restrict__ rdst,
                                  const int* __restrict__ src, const int* __restrict__ dst,
                                  const float* __restrict__ attn,
                                  float* __restrict__ alpha, unsigned* __restrict__ mord) {
    int tid = blockIdx.x * blockDim.x + threadIdx.x;
    int e = tid >> 4;
    if (e >= NEDGE) return;
    int t = tid & 15;
    int s = src[e], d = dst[e];
    float4 xi = ((const float4*)(rdst + (size_t)d * D_OUT))[t];
    float4 xj = ((const float4*)(lsrc + (size_t)s * D_OUT))[t];
    int h  = t >> 2;
    int cb = (t & 3) * 4;               // channel offset within head
    const float* aw = attn + h * (2 * C);
    float p = 0.f;
    p += aw[cb + 0] * lrelu(xi.x);
    p += aw[cb + 1] * lrelu(xi.y);
    p += aw[cb + 2] * lrelu(xi.z);
    p += aw[cb + 3] * lrelu(xi.w);
    p += aw[C + cb + 0] * lrelu(xj.x);
    p += aw[C + cb + 1] * lrelu(xj.y);
    p += aw[C + cb + 2] * lrelu(xj.z);
    p += aw[C + cb + 3] * lrelu(xj.w);
    p += __shfl_xor(p, 1, 32);
    p += __shfl_xor(p, 2, 32);
    if ((t & 3) == 0) {
        alpha[(size_t)e * H + h] = p;
        atomicMax(&mord[d * H + h], f2o(p));
    }
}

// ---------------------------------------------------------------------------
// Pass 2: ea = exp(alpha - m), denom += ea, acc += x_j * ea (L2 float atomics).
// Final emb = acc / (denom + 1e-16) == sum(x_j * ea/denom)  (same as reference).
// ---------------------------------------------------------------------------
__global__ void edge_accum_kernel(const float* __restrict__ lsrc,
                                  const int* __restrict__ src, const int* __restrict__ dst,
                                  const float* __restrict__ alpha, const unsigned* __restrict__ mord,
                                  float* __restrict__ denom, float* __restrict__ acc) {
    int tid = blockIdx.x * blockDim.x + threadIdx.x;
    int e = tid >> 4;
    if (e >= NEDGE) return;
    int t = tid & 15;
    int h = t >> 2;
    int s = src[e], d = dst[e];
    float m  = o2f(mord[d * H + h]);
    float ea = __expf(alpha[(size_t)e * H + h] - m);
    if ((t & 3) == 0) atomicAdd(&denom[d * H + h], ea);
    float4 xj = ((const float4*)(lsrc + (size_t)s * D_OUT))[t];
    float* ap = acc + (size_t)d * D_OUT + t * 4;
    atomicAdd(ap + 0, xj.x * ea);
    atomicAdd(ap + 1, xj.y * ea);
    atomicAdd(ap + 2, xj.z * ea);
    atomicAdd(ap + 3, xj.w * ea);
}

// ---------------------------------------------------------------------------
// Finalize: semantic (3-relation) softmax attention + ReLU. Thread = (n, h).
// ---------------------------------------------------------------------------
__global__ void finalize_kernel(const float* __restrict__ acc_w, const float* __restrict__ den_w,
                                const float* __restrict__ acc_c, const float* __restrict__ den_c,
                                const float* __restrict__ l_paper,
                                const float* __restrict__ rel_l, const float* __restrict__ rel_r,
                                float* __restrict__ out) {
    int tid = blockIdx.x * blockDim.x + threadIdx.x;
    if (tid >= N_OUT * H) return;
    int n = tid >> 2, h = tid & 3;
    float dw = den_w[n * H + h] + 1e-16f;
    float dc = den_c[n * H + h] + 1e-16f;
    const float* awp = acc_w   + (size_t)n * D_OUT + h * C;
    const float* acp = acc_c   + (size_t)n * D_OUT + h * C;
    const float* lpp = l_paper + (size_t)n * D_OUT + h * C;
    const float* rl = rel_l + h * C;
    const float* r0 = rel_r + (0 * H + h) * C;
    const float* r1 = rel_r + (1 * H + h) * C;
    const float* r2 = rel_r + (2 * H + h) * C;
    float ew[C], ec[C], es[C];
    float al = 0.f, a0 = 0.f, a1 = 0.f, a2 = 0.f;
#pragma unroll
    for (int c = 0; c < C; ++c) {
        ew[c] = awp[c] / dw;
        ec[c] = acp[c] / dc;
        es[c] = lpp[c];
        al += es[c] * rl[c];
        a0 += ew[c] * r0[c];
        a1 += ec[c] * r1[c];
        a2 += es[c] * r2[c];
    }
    float b0 = lrelu(al + a0), b1 = lrelu(al + a1), b2 = lrelu(al + a2);
    float mx = fmaxf(b0, fmaxf(b1, b2));
    float e0 = __expf(b0 - mx), e1 = __expf(b1 - mx), e2 = __expf(b2 - mx);
    float inv = 1.f / (e0 + e1 + e2);
    e0 *= inv; e1 *= inv; e2 *= inv;
    float* op = out + (size_t)n * D_OUT + h * C;
#pragma unroll
    for (int c = 0; c < C; ++c) {
        float v = ew[c] * e0 + ec[c] * e1 + es[c] * e2;
        op[c] = v > 0.f ? v : 0.f;
    }
}

// ---------------------------------------------------------------------------
extern "C" void kernel_launch(void* const* d_in, const int* in_sizes, int n_in,
                              void* d_out, int out_size, void* d_ws, size_t ws_size,
                              hipStream_t stream) {
    (void)in_sizes; (void)n_in; (void)out_size; (void)ws_size;
    const float* x_author   = (const float*)d_in[0];
    const float* x_paper    = (const float*)d_in[1];
    const float* W_l_author = (const float*)d_in[2];
    const float* b_l_author = (const float*)d_in[3];
    const float* W_l_paper  = (const float*)d_in[4];
    const float* b_l_paper  = (const float*)d_in[5];
    const float* W_r_paper  = (const float*)d_in[6];
    const float* b_r_paper  = (const float*)d_in[7];
    const float* attn_writes = (const float*)d_in[8];
    const float* attn_cites  = (const float*)d_in[9];
    const float* rel_attn_l  = (const float*)d_in[10];
    const float* rel_attn_r  = (const float*)d_in[11];
    const int* src_writes = (const int*)d_in[12];
    const int* dst_writes = (const int*)d_in[13];
    const int* src_cites  = (const int*)d_in[14];
    const int* dst_cites  = (const int*)d_in[15];
    float* out = (float*)d_out;

    // Workspace layout (bytes), all 16B-aligned; total ~163.2 MB.
    char* ws = (char*)d_ws;
    size_t off = 0;
    float*    l_author = (float*)(ws + off);    off += (size_t)N_AUTHOR * D_OUT * 4;  // 38.4 MB
    float*    l_paper  = (float*)(ws + off);    off += (size_t)N_PAPER  * D_OUT * 4;  // 51.2 MB
    float*    r_paper  = (float*)(ws + off);    off += (size_t)N_OUT    * D_OUT * 4;  // 12.8 MB
    float*    alpha_w  = (float*)(ws + off);    off += (size_t)NEDGE * H * 4;         // 16 MB
    float*    alpha_c  = (float*)(ws + off);    off += (size_t)NEDGE * H * 4;         // 16 MB
    unsigned* mord_w   = (unsigned*)(ws + off); off += (size_t)N_OUT * H * 4;
    unsigned* mord_c   = (unsigned*)(ws + off); off += (size_t)N_OUT * H * 4;
    float*    denom_w  = (float*)(ws + off);    off += (size_t)N_OUT * H * 4;
    float*    denom_c  = (float*)(ws + off);    off += (size_t)N_OUT * H * 4;
    float*    acc_w    = (float*)(ws + off);    off += (size_t)N_OUT * D_OUT * 4;     // 12.8 MB
    float*    acc_c    = (float*)(ws + off);    off += (size_t)N_OUT * D_OUT * 4;     // 12.8 MB

    // Dense projections via fp32 WMMA. 256 blocks x 8 waves = 2048 waves
    // (512 M-tile streams per 16-wide N tile).
    gemm_bias_wmma<<<256, 256, 0, stream>>>(x_author, W_l_author, b_l_author, l_author, N_AUTHOR / 16);
    gemm_bias_wmma<<<256, 256, 0, stream>>>(x_paper,  W_l_paper,  b_l_paper,  l_paper,  N_PAPER  / 16);
    gemm_bias_wmma<<<256, 256, 0, stream>>>(x_paper,  W_r_paper,  b_r_paper,  r_paper,  N_OUT    / 16);

    init_kernel<<<(N_OUT * D_OUT + 255) / 256, 256, 0, stream>>>(mord_w, mord_c, denom_w, denom_c, acc_w, acc_c);

    const int edgeThreads = NEDGE * 16;
    const int edgeBlocks  = (edgeThreads + 255) / 256;
    edge_alpha_kernel<<<edgeBlocks, 256, 0, stream>>>(l_author, r_paper, src_writes, dst_writes,
                                                      attn_writes, alpha_w, mord_w);
    edge_alpha_kernel<<<edgeBlocks, 256, 0, stream>>>(l_paper, r_paper, src_cites, dst_cites,
                                                      attn_cites, alpha_c, mord_c);
    edge_accum_kernel<<<edgeBlocks, 256, 0, stream>>>(l_author, src_writes, dst_writes,
                                                      alpha_w, mord_w, denom_w, acc_w);
    edge_accum_kernel<<<edgeBlocks, 256, 0, stream>>>(l_paper, src_cites, dst_cites,
                                                      alpha_c, mord_c, denom_c, acc_c);

    finalize_kernel<<<(N_OUT * H + 255) / 256, 256, 0, stream>>>(acc_w, denom_w, acc_c, denom_c,
                                                                 l_paper, rel_attn_l, rel_attn_r, out);
}